// SingleActionGNNPolicy_12463995093093
// MI455X (gfx1250) — compile-verified
//
#include <hip/hip_runtime.h>
#include <hip/hip_bf16.h>

typedef __attribute__((ext_vector_type(2))) float v2f;
typedef __attribute__((ext_vector_type(8))) float v8f;

// Order-preserving int encoding of float for atomicMax (no NaNs expected).
__device__ __forceinline__ int enc_ord(float f) {
    int i = __float_as_int(f);
    return (i < 0) ? (i ^ 0x7fffffff) : i;
}
__device__ __forceinline__ float dec_ord(int o) {
    int i = (o < 0) ? (o ^ 0x7fffffff) : o;
    return __int_as_float(i);
}

__global__ void init_accum_kernel(int* seg_max_o, float* seg_sum, float* seg_t, int B) {
    int g = blockIdx.x * blockDim.x + threadIdx.x;
    if (g < B) {
        seg_max_o[g] = enc_ord(-__builtin_huge_valf());
        seg_sum[g] = 0.0f;
        seg_t[g]   = 0.0f;
    }
}

// Pass 1: streaming matvec via V_WMMA_F32_16X16X4_F32.
// One wave computes logits for 16 consecutive nodes (rows of h).
// A (16x4 f32): lane L (<16) holds h[row L][4k+0..1]; lane L+16 holds h[row L][4k+2..3].
// B (4x16 f32): every column n broadcasts W[4k+0..3] (same per-lane addressing as A, from W).
// After 32 accumulating WMMAs, D[m,n] == dot(h[m], W) for every n.
__global__ void logits_wmma_kernel(const float* __restrict__ h,
                                   const float* __restrict__ W,
                                   const float* __restrict__ bias,
                                   const unsigned char* __restrict__ node_mask,
                                   const int* __restrict__ batch_idx,
                                   float* __restrict__ logits,
                                   int* __restrict__ seg_max_o,
                                   int N) {
    const int lane = threadIdx.x & 31;
    const int wave = threadIdx.x >> 5;
    const long tile = (long)blockIdx.x * 8 + wave;   // 16 nodes per tile
    const long base = tile * 16;

    const int  half = lane >> 4;                      // 0: K 0..1, 1: K 2..3
    long row = base + (lane & 15);
    if (row > (long)N - 1) row = (long)N - 1;         // clamp (keep EXEC all-1 for WMMA)
    const float* hrow = h + row * 128;
    const float* wptr = W + half * 2;

    v8f acc = {};
#pragma unroll
    for (int k = 0; k < 32; ++k) {
        v2f a = *(const v2f*)(hrow + k * 4 + half * 2);
        v2f bv = *(const v2f*)(wptr + k * 4);
        // 8 args: (neg_a, A, neg_b, B, c_mod, C, reuse_a, reuse_b)
        acc = __builtin_amdgcn_wmma_f32_16x16x4_f32(false, a, false, bv,
                                                    (short)0, acc, false, false);
    }

    // Extract: node m of this tile lives in D-vgpr (m&7), lanes 0-15 if m<8 else 16-31.
    const int j = lane & 7;
    float v01 = (j & 1) ? acc[1] : acc[0];
    float v23 = (j & 1) ? acc[3] : acc[2];
    float v45 = (j & 1) ? acc[5] : acc[4];
    float v67 = (j & 1) ? acc[7] : acc[6];
    float v03 = (j & 2) ? v23 : v01;
    float v47 = (j & 2) ? v67 : v45;
    float dot = (j & 4) ? v47 : v03;

    if ((lane & 8) == 0) {
        const int m = ((lane >> 4) << 3) | j;        // 0..15
        const long node = base + m;
        if (node < N) {
            float logit = dot + bias[0];
            float masked = node_mask[node] ? logit : -1e30f;
            logits[node] = masked;
            atomicMax(&seg_max_o[batch_idx[node]], enc_ord(masked));
        }
    }
}

// Pass 2: ex = exp(x - M); accumulate S = sum(ex), T = sum(ex*(x - M)) per graph.
// batch_idx is sorted, so waves are almost always graph-uniform: pre-reduce in-wave.
__global__ void segsum_kernel(const float* __restrict__ logits,
                              const int* __restrict__ batch_idx,
                              const unsigned char* __restrict__ node_mask,
                              const int* __restrict__ seg_max_o,
                              float* __restrict__ seg_sum,
                              float* __restrict__ seg_t,
                              int N) {
    const int n = blockIdx.x * blockDim.x + threadIdx.x;
    const int lane = threadIdx.x & 31;
    const int nn = (n < N) ? n : (N - 1);

    const int g = batch_idx[nn];
    float ex = 0.0f, t = 0.0f;
    if (n < N && node_mask[nn]) {
        float M = dec_ord(seg_max_o[g]);
        float d = logits[nn] - M;
        ex = expf(d);
        t = ex * d;
    }

    const bool uniform = __all(g == __shfl(g, 0, 32));
    if (uniform) {
        for (int off = 16; off > 0; off >>= 1) {
            ex += __shfl_xor(ex, off, 32);
            t  += __shfl_xor(t, off, 32);
        }
        if (lane == 0 && (ex != 0.0f || t != 0.0f)) {
            atomicAdd(&seg_sum[g], ex);
            atomicAdd(&seg_t[g], t);
        }
    } else {
        if (ex != 0.0f || t != 0.0f) {
            atomicAdd(&seg_sum[g], ex);
            atomicAdd(&seg_t[g], t);
        }
    }
}

// Pass 3: entropy[g] = log S - T/S ;  logprob[b] = log( exp(x_a - M_g)/max(S_g,1e-30) )
__global__ void finalize_kernel(const int* __restrict__ actions,
                                const int* __restrict__ batch_idx,
                                const float* __restrict__ logits,
                                const int* __restrict__ seg_max_o,
                                const float* __restrict__ seg_sum,
                                const float* __restrict__ seg_t,
                                float* __restrict__ out_logprob,
                                float* __restrict__ out_entropy,
                                int B) {
    const int b = blockIdx.x * blockDim.x + threadIdx.x;
    if (b >= B) return;

    float S = seg_sum[b];
    float T = seg_t[b];
    out_entropy[b] = (S > 0.0f) ? (logf(fmaxf(S, 1e-30f)) - T / S) : 0.0f;

    int a = actions[b];
    int g = batch_idx[a];
    float M  = dec_ord(seg_max_o[g]);
    float Sg = fmaxf(seg_sum[g], 1e-30f);
    float p  = expf(logits[a] - M) / Sg;   // masked nodes stored -1e30 -> exp underflows to 0
    out_logprob[b] = logf(p);
}

extern "C" void kernel_launch(void* const* d_in, const int* in_sizes, int n_in,
                              void* d_out, int out_size, void* d_ws, size_t ws_size,
                              hipStream_t stream) {
    const int*           actions   = (const int*)d_in[0];
    const float*         h         = (const float*)d_in[1];
    const int*           batch_idx = (const int*)d_in[2];
    const unsigned char* node_mask = (const unsigned char*)d_in[3];
    // d_in[4] = n_graphs scalar (device); B derived host-side from actions size.
    const float*         W         = (const float*)d_in[5];
    const float*         bias      = (const float*)d_in[6];

    const int N = in_sizes[2];   // batch_idx length
    const int B = in_sizes[0];   // actions length == n_graphs

    float* out_logprob = (float*)d_out;
    float* out_entropy = out_logprob + B;

    // Workspace layout: logits[N] f32 | seg_max_o[B] i32 | seg_sum[B] f32 | seg_t[B] f32
    float* logits    = (float*)d_ws;
    int*   seg_max_o = (int*)(logits + N);
    float* seg_sum   = (float*)(seg_max_o + B);
    float* seg_t     = seg_sum + B;

    init_accum_kernel<<<(B + 255) / 256, 256, 0, stream>>>(seg_max_o, seg_sum, seg_t, B);

    const int tiles  = (N + 15) / 16;          // 16 nodes per wave
    const int blocks = (tiles + 7) / 8;        // 8 waves per 256-thread block
    logits_wmma_kernel<<<blocks, 256, 0, stream>>>(h, W, bias, node_mask, batch_idx,
                                                   logits, seg_max_o, N);

    segsum_kernel<<<(N + 255) / 256, 256, 0, stream>>>(logits, batch_idx, node_mask,
                                                       seg_max_o, seg_sum, seg_t, N);

    finalize_kernel<<<(B + 255) / 256, 256, 0, stream>>>(actions, batch_idx, logits,
                                                         seg_max_o, seg_sum, seg_t,
                                                         out_logprob, out_entropy, B);
}